// EMASmoothCUDA_79972291052168
// MI455X (gfx1250) — compile-verified
//
#include <hip/hip_runtime.h>
#include <math.h>

// ---------------------------------------------------------------------------
// EMASmooth: out = x + g * (ema - x),  ema_t = a*ema_{t-1} + (1-a)*x_t
// B=8, T=4096, D=1024, f32.  Bandwidth-bound (~268MB => ~11.5us @ 23.3TB/s).
// Chunked-scan decomposition: T split into 32 chunks of 128; carries combined
// with decay alpha^128 (exact: 7 squarings).  Pass 2 streams tiles via the
// CDNA5 Tensor Data Mover, double-buffered with s_wait_tensorcnt.
// ---------------------------------------------------------------------------

#define BB      8
#define TT      4096
#define DD      1024
#define NCHUNK  32
#define CHUNK   128          // TT / NCHUNK
#define HCHUNK  64           // half-chunk rows per TDM transfer
#define DTILE   256          // channels per block == blockDim.x

typedef uint __attribute__((ext_vector_type(4))) v4u;
typedef int  __attribute__((ext_vector_type(4))) v4i;
typedef int  __attribute__((ext_vector_type(8))) v8i;

#define HAS_TDM (__has_builtin(__builtin_amdgcn_tensor_load_to_lds) && \
                 __has_builtin(__builtin_amdgcn_s_wait_tensorcnt))

__device__ __forceinline__ float sigmoidf_(float z) {
    float e = __expf(-z);
#if __has_builtin(__builtin_amdgcn_rcpf)
    return __builtin_amdgcn_rcpf(1.0f + e);   // v_rcp_f32, ~1ulp
#else
    return 1.0f / (1.0f + e);
#endif
}

#if HAS_TDM
// Issue one TDM 2-D tile load: `rows` rows of DTILE f32 elements, global row
// stride DD elements, into LDS at byte offset lds_off.  Uniform args only.
__device__ __forceinline__ void tdm_load_tile(const float* gptr, unsigned lds_off,
                                              unsigned rows) {
    const unsigned long long ga = (unsigned long long)(uintptr_t)gptr;

    // ---- D# group 0 (128b): count=1 | lds_addr | global_addr[56:0] | type=2
    v4u g0;
    g0[0] = 1u;                                    // count=1, gather off
    g0[1] = lds_off;                               // lds_addr (bytes)
    g0[2] = (unsigned)(ga & 0xFFFFFFFFull);        // global_addr[31:0]
    g0[3] = (unsigned)((ga >> 32) & 0x01FFFFFFull) | (2u << 30); // [56:32] | type=2

    // ---- D# group 1 (256b): 2-D tile, 4-byte elements
    const unsigned td0  = 0x40000000u;             // tensor_dim0 (large: no OOB)
    const unsigned td1  = 0x40000000u;             // tensor_dim1 (large: no OOB)
    const unsigned tld0 = DTILE;                   // tile_dim0 = 256 elems/row
    const unsigned long long st0 = DD;             // tensor_dim0_stride = 1024 elems
    v8i g1;
    g1[0] = (int)(2u << 16);                              // data_size = 4B
    g1[1] = (int)((td0 & 0xFFFFu) << 16);                 // td0[15:0]
    g1[2] = (int)(((td1 & 0xFFFFu) << 16) | (td0 >> 16)); // td1[15:0] | td0[31:16]
    g1[3] = (int)((tld0 << 16) | (td1 >> 16));            // tile_dim0 | td1[31:16]
    g1[4] = (int)rows;                                    // tile_dim2=0 | tile_dim1
    g1[5] = (int)(unsigned)(st0 & 0xFFFFFFFFull);         // stride0[31:0]
    g1[6] = (int)(unsigned)((st0 >> 32) & 0xFFFFull);     // s1[15:0]=0 | stride0[47:32]
    g1[7] = 0;                                            // stride1[47:16] = 0

    v4i g2 = {0, 0, 0, 0};                // 2-D tile: dims 2/3 unused
    v4i g3 = {0, 0, 0, 0};
    v8i g4 = {0, 0, 0, 0, 0, 0, 0, 0};    // extended slot (unused)

    __builtin_amdgcn_tensor_load_to_lds(g0, g1, g2, g3, g4, 0);
}
#endif

// ---------------------------------------------------------------------------
// Pass 1: per-chunk local EMA with zero init -> carry[b][chunk][d]
// ---------------------------------------------------------------------------
__global__ void ema_pass1(const float* __restrict__ x,
                          const float* __restrict__ alpha_logit,
                          float* __restrict__ carry) {
    const int d = blockIdx.x * DTILE + threadIdx.x;
    const int c = blockIdx.y;
    const int b = blockIdx.z;

    const float alpha = sigmoidf_(alpha_logit[d]);
    const float onem  = 1.0f - alpha;

    size_t base = ((size_t)b * TT + (size_t)c * CHUNK) * DD + d;
    float ema = 0.0f;
#pragma unroll 8
    for (int r = 0; r < CHUNK; ++r) {
        float xv = x[base + (size_t)r * DD];
        ema = fmaf(alpha, ema, onem * xv);
    }
    carry[((size_t)b * NCHUNK + c) * DD + d] = ema;
}

// ---------------------------------------------------------------------------
// Combine: exclusive scan of carries along chunks with decay alpha^CHUNK.
// carry_in[b][c][d] = EMA state entering chunk c.
// ---------------------------------------------------------------------------
__global__ void ema_combine(const float* __restrict__ carry,
                            const float* __restrict__ alpha_logit,
                            float* __restrict__ carry_in) {
    const int idx = blockIdx.x * blockDim.x + threadIdx.x;   // [0, B*D)
    const int b = idx >> 10;          // /1024
    const int d = idx & (DD - 1);

    float alpha = sigmoidf_(alpha_logit[d]);
    // alpha^128 exactly: 7 squarings
    float aL = alpha;
#pragma unroll
    for (int i = 0; i < 7; ++i) aL *= aL;

    float E = 0.0f;
#pragma unroll 4
    for (int k = 0; k < NCHUNK; ++k) {
        size_t off = ((size_t)b * NCHUNK + k) * DD + d;
        carry_in[off] = E;
        E = fmaf(aL, E, carry[off]);
    }
}

// ---------------------------------------------------------------------------
// Pass 2: stream each (128 t x 256 d) tile global->LDS with the TDM
// (double-buffered halves), finish the scan with the correct incoming carry,
// apply the gate, write out with coalesced streaming stores.
// ---------------------------------------------------------------------------
__global__ void ema_pass2(const float* __restrict__ x,
                          const float* __restrict__ alpha_logit,
                          const float* __restrict__ gate_logit,
                          const float* __restrict__ carry_in,
                          float* __restrict__ out) {
    extern __shared__ float smem[];           // CHUNK * DTILE floats = 128 KB
    const int tid = threadIdx.x;
    const int d   = blockIdx.x * DTILE + tid;
    const int c   = blockIdx.y;
    const int b   = blockIdx.z;

    // element index of tile origin: x[b, c*CHUNK, blockIdx.x*DTILE]
    const size_t tile0 = ((size_t)b * TT + (size_t)c * CHUNK) * DD
                       + (size_t)blockIdx.x * DTILE;

    const float alpha = sigmoidf_(alpha_logit[d]);
    const float onem  = 1.0f - alpha;
    const float g     = sigmoidf_(gate_logit[0]);
    float ema = carry_in[((size_t)b * NCHUNK + c) * DD + d];
    const size_t gbase = tile0 + tid;

#if HAS_TDM
    const unsigned lds0 = (unsigned)(uintptr_t)smem;   // low 32 bits = LDS offset
    if (tid < 32) {   // one wave drives the DMA; TENSORcnt is per-wave
        tdm_load_tile(x + tile0, lds0, HCHUNK);                            // half 0
        tdm_load_tile(x + tile0 + (size_t)HCHUNK * DD,
                      lds0 + HCHUNK * DTILE * sizeof(float), HCHUNK);      // half 1
        __builtin_amdgcn_s_wait_tensorcnt((short)1);   // half 0 landed (in-order)
    }
    __syncthreads();

    // ---- compute half 0 while half 1 DMA is still in flight
#pragma unroll 8
    for (int r = 0; r < HCHUNK; ++r) {
        float xv = smem[r * DTILE + tid];
        ema = fmaf(alpha, ema, onem * xv);
        out[gbase + (size_t)r * DD] = fmaf(g, ema - xv, xv);
    }

    if (tid < 32) __builtin_amdgcn_s_wait_tensorcnt((short)0);
    __syncthreads();

    // ---- compute half 1
#pragma unroll 8
    for (int r = HCHUNK; r < CHUNK; ++r) {
        float xv = smem[r * DTILE + tid];
        ema = fmaf(alpha, ema, onem * xv);
        out[gbase + (size_t)r * DD] = fmaf(g, ema - xv, xv);
    }
#else
    // Fallback: cooperative coalesced fill of the LDS tile
    for (int i = tid; i < CHUNK * DTILE; i += blockDim.x) {
        int r  = i >> 8;            // /DTILE
        int dd = i & (DTILE - 1);
        smem[i] = x[tile0 + (size_t)r * DD + dd];
    }
    __syncthreads();
#pragma unroll 8
    for (int r = 0; r < CHUNK; ++r) {
        float xv = smem[r * DTILE + tid];
        ema = fmaf(alpha, ema, onem * xv);
        out[gbase + (size_t)r * DD] = fmaf(g, ema - xv, xv);
    }
#endif
}

// ---------------------------------------------------------------------------
extern "C" void kernel_launch(void* const* d_in, const int* in_sizes, int n_in,
                              void* d_out, int out_size, void* d_ws, size_t ws_size,
                              hipStream_t stream) {
    const float* x           = (const float*)d_in[0];   // [B,T,D]
    const float* alpha_logit = (const float*)d_in[1];   // [D]
    const float* gate_logit  = (const float*)d_in[2];   // [1]
    float*       out         = (float*)d_out;

    float* carry    = (float*)d_ws;                         // B*NCHUNK*D floats (1MB)
    float* carry_in = carry + (size_t)BB * NCHUNK * DD;     // another 1MB

    const int smem_bytes = CHUNK * DTILE * sizeof(float);   // 128 KB (LDS is 320KB/WGP)
    (void)hipFuncSetAttribute((const void*)ema_pass2,
                              hipFuncAttributeMaxDynamicSharedMemorySize, smem_bytes);

    dim3 grid1(DD / DTILE, NCHUNK, BB);   // 4 x 32 x 8 blocks
    dim3 block(DTILE);                    // 256 threads = 8 wave32

    ema_pass1<<<grid1, block, 0, stream>>>(x, alpha_logit, carry);

    dim3 grid2((BB * DD) / DTILE);        // 32 blocks
    ema_combine<<<grid2, block, 0, stream>>>(carry, alpha_logit, carry_in);

    ema_pass2<<<grid1, block, smem_bytes, stream>>>(x, alpha_logit, gate_logit,
                                                    carry_in, out);
}